// StreamingRDA_4269197492338
// MI455X (gfx1250) — compile-verified
//
#include <hip/hip_runtime.h>
#include <hip/hip_bf16.h>
#include <math.h>

// Problem constants (match reference)
#define Dsz     512
#define Ccls    100
#define Nsmp    2048
#define SHRINKF 1e-4f
#define NS_ITERS 16

#define BM 64
#define BN 64
#define BK 16

typedef __attribute__((ext_vector_type(2))) float v2f;
typedef __attribute__((ext_vector_type(8))) float v8f;
typedef unsigned int u32x4 __attribute__((ext_vector_type(4)));
typedef int          i32x8 __attribute__((ext_vector_type(8)));
typedef int          i32x4 __attribute__((ext_vector_type(4)));

static __device__ __forceinline__ v8f wmma4(v2f a, v2f b, v8f c) {
  // D = A(16x4) * B(4x16) + C, fp32 everywhere -> v_wmma_f32_16x16x4_f32
  return __builtin_amdgcn_wmma_f32_16x16x4_f32(false, a, false, b, (short)0, c,
                                               false, false);
}

static __device__ __forceinline__ v8f v8zero() {
  v8f z;
#pragma unroll
  for (int i = 0; i < 8; ++i) z[i] = 0.0f;
  return z;
}

// ---- Tensor Data Mover: DMA a 2D fp32 tile (tileY rows x tileX cols, row
// stride strideX elements) from global into LDS at byte offset ldsOff.
// padCode: pad_interval encoding (3 = pad 1 dword every 16 dwords -> LDS
// stride 17; 5 = every 64 dwords -> stride 65). ISA cdna5 ch.8 (D# layout).
// This toolchain exposes the 6-arg tensor_load_to_lds builtin
// (g0 v4u, g1 v8i, g2 v4i, g3 v4i, extra v8i, cpol).
static __device__ __forceinline__ void tdm_load_2d(unsigned ldsOff,
                                                   const float* g,
                                                   int tileX, int tileY,
                                                   int strideX, int padCode) {
  unsigned long long ga = (unsigned long long)(uintptr_t)g;
  u32x4 g0;
  g0[0] = 1u;                                   // count=1 (valid), user mode
  g0[1] = ldsOff;                               // lds_addr (bytes)
  g0[2] = (unsigned)(ga & 0xFFFFFFFFu);         // global_addr[31:0]
  g0[3] = (unsigned)((ga >> 32) & 0x1FFFFFFu)   // global_addr[56:32]
          | (2u << 30);                         // type = 2 ("image")
  i32x8 g1;
  unsigned dw0 = (2u << 16)                     // data_size = 4 bytes
               | (1u << 20)                     // pad_enable
               | ((unsigned)padCode << 22);     // pad_interval (pad_amount=0 -> 1 dword)
  g1[0] = (int)dw0;
  g1[1] = (int)(((unsigned)strideX & 0xFFFFu) << 16);   // tensor_dim0[15:0]
  g1[2] = (int)((((unsigned)strideX >> 16) & 0xFFFFu)   // tensor_dim0[31:16]
                | (0xFFFFu << 16));                     // tensor_dim1[15:0] = big
  g1[3] = (int)(((unsigned)tileX & 0xFFFFu) << 16);     // tensor_dim1 hi=0 | tile_dim0
  g1[4] = (int)((unsigned)tileY & 0xFFFFu);             // tile_dim1 (tile_dim2=0)
  g1[5] = (int)(unsigned)strideX;                       // tensor_dim0_stride[31:0]
  g1[6] = 0;                                            // stride hi / dim1_stride
  g1[7] = 0;
  i32x4 z4; z4[0] = 0; z4[1] = 0; z4[2] = 0; z4[3] = 0;
  i32x8 z8;
#pragma unroll
  for (int i = 0; i < 8; ++i) z8[i] = 0;
  __builtin_amdgcn_tensor_load_to_lds(g0, g1, z4, z4, z8, 0);
}

// ---------------- prep kernels ----------------

// M[k] = (1-sh) * (0.5*SigmaK[k] + 0.5*Sigma) + sh * I
__global__ __launch_bounds__(256) void prep_M_kernel(
    const float* __restrict__ SigK, const float* __restrict__ Sig,
    float* __restrict__ M)
{
  size_t idx = (size_t)blockIdx.x * 256 + threadIdx.x;  // < C*D*D (exact)
  size_t rem = idx % ((size_t)Dsz * Dsz);
  int r = (int)(rem / Dsz), c = (int)(rem % Dsz);
  float reg = 0.5f * SigK[idx] + 0.5f * Sig[rem];
  M[idx] = (1.0f - SHRINKF) * reg + ((r == c) ? SHRINKF : 0.0f);
}

// Per class: halflog[k] = 0.5*log(||Reg||_F) = 0.25*log(sum Reg^2)
//            invscale[k] = 1 / max_row_abs_sum(M)   (Newton-Schulz init)
__global__ __launch_bounds__(256) void class_stats_kernel(
    const float* __restrict__ SigK, const float* __restrict__ Sig,
    float* __restrict__ halflog, float* __restrict__ invscale)
{
  int k = blockIdx.x, t = threadIdx.x;
  const float* Sk = SigK + (size_t)k * Dsz * Dsz;
  float fn2 = 0.0f, rmax = 0.0f;
  for (int r = t; r < Dsz; r += 256) {
    float rs = 0.0f;
    for (int c = 0; c < Dsz; ++c) {
      float reg = 0.5f * Sk[(size_t)r * Dsz + c] + 0.5f * Sig[(size_t)r * Dsz + c];
      fn2 += reg * reg;
      float m = (1.0f - SHRINKF) * reg + ((r == c) ? SHRINKF : 0.0f);
      rs += fabsf(m);
    }
    rmax = fmaxf(rmax, rs);
  }
  __shared__ float sfn[256], smx[256];
  sfn[t] = fn2; smx[t] = rmax;
  __syncthreads();
  for (int s = 128; s > 0; s >>= 1) {
    if (t < s) { sfn[t] += sfn[t + s]; smx[t] = fmaxf(smx[t], smx[t + s]); }
    __syncthreads();
  }
  if (t == 0) {
    halflog[k]  = 0.25f * logf(sfn[0]);
    invscale[k] = 1.0f / smx[0];
  }
}

__global__ __launch_bounds__(128) void prior_kernel(
    const float* __restrict__ cK, float* __restrict__ logprior)
{
  __shared__ float ssum[128];
  int t = threadIdx.x;
  float s = 0.0f;
  for (int i = t; i < Ccls; i += 128) s += cK[i];
  ssum[t] = s;
  __syncthreads();
  for (int st = 64; st > 0; st >>= 1) {
    if (t < st) ssum[t] += ssum[t + st];
    __syncthreads();
  }
  float tot = ssum[0];
  if (t < Ccls) {
    float ck = cK[t];
    float pr = (ck == 0.0f) ? 1.0f : (ck / tot);
    logprior[t] = logf(pr);
  }
}

// X0[k] = invscale[k] * I
__global__ __launch_bounds__(256) void init_X_kernel(
    const float* __restrict__ invscale, float* __restrict__ X0)
{
  size_t idx = (size_t)blockIdx.x * 256 + threadIdx.x;
  size_t DD = (size_t)Dsz * Dsz;
  int k = (int)(idx / DD);
  size_t rem = idx % DD;
  int r = (int)(rem / Dsz), c = (int)(rem % Dsz);
  X0[idx] = (r == c) ? invscale[k] : 0.0f;
}

// ---------------- batched WMMA GEMM: Out[k] = alpha*A[k]@B[k] + beta*Cin[k] ----------------
// 512x512x512 per class; block = 128 threads (4 waves), 64x64 macro-tile.
// Tiles staged into LDS by the Tensor Data Mover (wave 0 issues; TENSORcnt + barrier).
__global__ __launch_bounds__(128) void gemm_wmma_kernel(
    const float* __restrict__ A, const float* __restrict__ B,
    const float* __restrict__ Cin, float* __restrict__ Out,
    float alpha, float beta)
{
  const size_t S = (size_t)Dsz * Dsz;
  int kc = blockIdx.z;
  const float* Ak = A + (size_t)kc * S;
  const float* Bk = B + (size_t)kc * S;
  const float* Ck = Cin + (size_t)kc * S;
  float* Ok = Out + (size_t)kc * S;

  __shared__ float As[BM][BK + 1];  // stride 17: conflict-free A-frag reads
  __shared__ float Bs[BK][BN + 1];  // stride 65
  unsigned asOff = (unsigned)(uintptr_t)&As[0][0];  // LDS byte offset (addr[31:0])
  unsigned bsOff = (unsigned)(uintptr_t)&Bs[0][0];

  int t = threadIdx.x;
  int wave = t >> 5, lane = t & 31, l16 = lane & 15, lhi = lane >> 4;
  int m0 = blockIdx.x * BM, n0 = blockIdx.y * BN;

  v8f acc[4];
#pragma unroll
  for (int i = 0; i < 4; ++i) acc[i] = v8zero();

  for (int k0 = 0; k0 < Dsz; k0 += BK) {
    __syncthreads();  // all waves done reading previous tiles
    if (wave == 0) {
      // A tile: 64 rows x 16 cols -> LDS stride 17 (pad 1 dword / 16)
      tdm_load_2d(asOff, Ak + (size_t)m0 * Dsz + k0, BK, BM, Dsz, 3);
      // B tile: 16 rows x 64 cols -> LDS stride 65 (pad 1 dword / 64)
      tdm_load_2d(bsOff, Bk + (size_t)k0 * Dsz + n0, BN, BK, Dsz, 5);
      __builtin_amdgcn_s_wait_tensorcnt(0);
    }
    __syncthreads();  // tiles visible to all waves

    int arow = wave * 16 + l16;
#pragma unroll
    for (int kk = 0; kk < BK; kk += 4) {
      int kb = kk + lhi * 2;
      // A frag (16x4): lanes 0-15 -> K=kk,kk+1 ; lanes 16-31 -> K=kk+2,kk+3
      v2f a; a[0] = As[arow][kb]; a[1] = As[arow][kb + 1];
#pragma unroll
      for (int nt = 0; nt < 4; ++nt) {
        // B frag (4x16): v0 = rows kk (lo half) / kk+2 (hi half); v1 = +1
        v2f b; b[0] = Bs[kb][nt * 16 + l16]; b[1] = Bs[kb + 1][nt * 16 + l16];
        acc[nt] = wmma4(a, b, acc[nt]);
      }
    }
  }

  // C/D layout: VGPR j holds rows (j + 8*hi-half), col = lane&15
  int rowb = m0 + wave * 16 + lhi * 8;
  int colb = n0 + l16;
#pragma unroll
  for (int nt = 0; nt < 4; ++nt) {
#pragma unroll
    for (int j = 0; j < 8; ++j) {
      size_t off = (size_t)(rowb + j) * Dsz + (colb + nt * 16);
      float v = alpha * acc[nt][j];
      if (beta != 0.0f) v += beta * Ck[off];
      Ok[off] = v;
    }
  }
}

// ---------------- fused Mahalanobis quad + scores ----------------
// block = (64-row tile of X) x (class k): quad[n] = rowsum((Diff @ Lam_k) .* Diff)
__global__ __launch_bounds__(128) void quad_scores_kernel(
    const float* __restrict__ X, const float* __restrict__ muK,
    const float* __restrict__ Lam, const float* __restrict__ halflog,
    const float* __restrict__ logprior, const float* __restrict__ cK,
    float* __restrict__ scores)
{
  int kc = blockIdx.y;
  int nbase = blockIdx.x * 64;
  const float* Lk = Lam + (size_t)kc * Dsz * Dsz;
  const float* mk = muK + (size_t)kc * Dsz;

  __shared__ float Dt[64][BK + 1];   // Diff tile (n, d), manual (fuses X - mu)
  __shared__ float Ls[BK][BN + 1];   // Lambda tile (d, e), TDM-staged
  __shared__ float qrow[64];
  unsigned lsOff = (unsigned)(uintptr_t)&Ls[0][0];

  int t = threadIdx.x;
  int wave = t >> 5, lane = t & 31, l16 = lane & 15, lhi = lane >> 4;
  if (t < 64) qrow[t] = 0.0f;

  float qacc[8];
#pragma unroll
  for (int j = 0; j < 8; ++j) qacc[j] = 0.0f;

  for (int e0 = 0; e0 < Dsz; e0 += 64) {
    v8f acc[4];
#pragma unroll
    for (int i = 0; i < 4; ++i) acc[i] = v8zero();

    for (int d0 = 0; d0 < Dsz; d0 += BK) {
      __syncthreads();  // previous stage fully consumed
      if (wave == 0) {  // Lambda tile via TDM: 16 x 64, LDS stride 65
        tdm_load_2d(lsOff, Lk + (size_t)d0 * Dsz + e0, BN, BK, Dsz, 5);
      }
#pragma unroll
      for (int i = 0; i < 8; ++i) {  // Diff tile, all 128 threads
        int idx = t + i * 128;
        int r = idx >> 4, c = idx & 15;
        Dt[r][c] = X[(size_t)(nbase + r) * Dsz + (d0 + c)] - mk[d0 + c];
      }
      if (wave == 0) __builtin_amdgcn_s_wait_tensorcnt(0);
      __syncthreads();

      int arow = wave * 16 + l16;
#pragma unroll
      for (int kk = 0; kk < BK; kk += 4) {
        int kb = kk + lhi * 2;
        v2f a; a[0] = Dt[arow][kb]; a[1] = Dt[arow][kb + 1];
#pragma unroll
        for (int nt = 0; nt < 4; ++nt) {
          v2f b; b[0] = Ls[kb][nt * 16 + l16]; b[1] = Ls[kb + 1][nt * 16 + l16];
          acc[nt] = wmma4(a, b, acc[nt]);
        }
      }
    }

    // Y .* Diff, accumulated per lane into the 8 rows this lane owns
    int rowb = wave * 16 + lhi * 8;
#pragma unroll
    for (int nt = 0; nt < 4; ++nt) {
      int col = e0 + nt * 16 + l16;
#pragma unroll
      for (int j = 0; j < 8; ++j) {
        int row = rowb + j;
        float diff = X[(size_t)(nbase + row) * Dsz + col] - mk[col];
        qacc[j] += acc[nt][j] * diff;
      }
    }
  }

  int rowf = wave * 16 + lhi * 8;
#pragma unroll
  for (int j = 0; j < 8; ++j) atomicAdd(&qrow[rowf + j], qacc[j]);  // ds_add_f32
  __syncthreads();

  if (t < 64) {
    float ck = cK[kc];
    float sc = (ck == 0.0f) ? -INFINITY
                            : (logprior[kc] - halflog[kc] - 0.5f * qrow[t]);
    scores[(size_t)(nbase + t) * Ccls + kc] = sc;
  }
}

// ---------------- host launch ----------------

extern "C" void kernel_launch(void* const* d_in, const int* in_sizes, int n_in,
                              void* d_out, int out_size, void* d_ws, size_t ws_size,
                              hipStream_t stream) {
  const float* X    = (const float*)d_in[0];
  const float* muK  = (const float*)d_in[1];
  const float* SigK = (const float*)d_in[2];
  const float* Sig  = (const float*)d_in[3];
  const float* cK   = (const float*)d_in[4];
  float* scores = (float*)d_out;

  float* ws = (float*)d_ws;
  float* halflog  = ws;         // [C]
  float* logprior = ws + 128;   // [C]
  float* invscale = ws + 256;   // [C]
  const size_t S = (size_t)Ccls * Dsz * Dsz;
  float* M  = ws + 512;         // shrunk matrices   [C,D,D]
  float* X0 = M + S;            // Newton ping       [C,D,D]
  float* X1 = X0 + S;           // Newton pong       [C,D,D]
  float* T  = X1 + S;           // temp M@X          [C,D,D]

  const int nElem = Ccls * Dsz * Dsz;  // 26,214,400 (divisible by 256)

  prep_M_kernel<<<nElem / 256, 256, 0, stream>>>(SigK, Sig, M);
  class_stats_kernel<<<Ccls, 256, 0, stream>>>(SigK, Sig, halflog, invscale);
  prior_kernel<<<1, 128, 0, stream>>>(cK, logprior);
  init_X_kernel<<<nElem / 256, 256, 0, stream>>>(invscale, X0);

  // Newton-Schulz: X <- 2X - X(MX), all WMMA GEMMs
  dim3 g(Dsz / BM, Dsz / BN, Ccls);
  float* cur = X0; float* nxt = X1;
  for (int it = 0; it < NS_ITERS; ++it) {
    gemm_wmma_kernel<<<g, 128, 0, stream>>>(M, cur, T, T, 1.0f, 0.0f);      // T = M@X
    gemm_wmma_kernel<<<g, 128, 0, stream>>>(cur, T, cur, nxt, -1.0f, 2.0f); // X' = 2X - X@T
    float* tmp = cur; cur = nxt; nxt = tmp;
  }
  // NS_ITERS is even -> result in X0 (== cur)

  dim3 gq(Nsmp / 64, Ccls);
  quad_scores_kernel<<<gq, 128, 0, stream>>>(X, muK, cur, halflog, logprior, cK,
                                             scores);

  (void)in_sizes; (void)n_in; (void)out_size; (void)ws_size;
}